// ElementBinner_51840255262842
// MI455X (gfx1250) — compile-verified
//
#include <hip/hip_runtime.h>
#include <stdint.h>

#define BB      8
#define NN      12800
#define FF      128
#define NBINS   800      /* B * 10 * 10 */
#define MAXE    256
#define ROWS    4        /* feature rows per wave in the gather stage */

/* ------------------------------------------------------------------ */
/* Kernel 1: per-element bin id (searchsorted 'left' over interior     */
/* edges of linspace grids).                                           */
/* ------------------------------------------------------------------ */
__global__ __launch_bounds__(256) void binner_gid_kernel(
    const float* __restrict__ coords,          /* (B,N,3): eta, sin, cos */
    const unsigned char* __restrict__ mask,    /* (B,N) bool             */
    int* __restrict__ gids)                    /* (B*N)                  */
{
    int idx = blockIdx.x * 256 + threadIdx.x;
    if (idx >= BB * NN) return;

    float eta = coords[idx * 3 + 0];
    float sp  = coords[idx * 3 + 1];
    float cp  = coords[idx * 3 + 2];
    float phi = atan2f(sp, cp);

    int eid = 0, pid = 0;
#pragma unroll
    for (int k = 0; k < 9; ++k) {
        float ee = -4.0f + (float)k;                                   /* eta interior edges */
        float pe = -3.14159265358979323846f
                 + 0.62831853071795864769f * (float)(k + 1);           /* phi interior edges */
        eid += (ee < eta) ? 1 : 0;   /* side='left': count edges strictly below */
        pid += (pe < phi) ? 1 : 0;
    }
    int b = idx / NN;
    int g = b * 100 + eid * 10 + pid;
    gids[idx] = mask[idx] ? g : 0x7FFFFFFF;    /* sentinel = overflow/dropped */
}

/* ------------------------------------------------------------------ */
/* Kernel 2: one wave32 per bin; in-order chunk scan + ballot/prefix-  */
/* popcount gives the STABLE within-bin rank (matches stable argsort). */
/* ------------------------------------------------------------------ */
__global__ __launch_bounds__(256) void binner_scan_kernel(
    const int* __restrict__ gids,              /* (B*N)          */
    int* __restrict__ slot_src)                /* (NBINS*MAXE)   */
{
    int wave = blockIdx.x * 8 + (threadIdx.x >> 5);   /* bin id */
    int lane = threadIdx.x & 31;
    if (wave >= NBINS) return;

    int b = wave / 100;
    const int* g = gids + (size_t)b * NN;
    int base = 0;

    for (int n = 0; n < NN; n += 32) {                /* NN % 32 == 0 */
        int gg = g[n + lane];
        bool match = (gg == wave);
        unsigned m = (unsigned)__ballot(match);       /* wave32: low 32 bits */
        if (match) {
            int slot = base + __popc(m & ((1u << lane) - 1u));
            if (slot < MAXE)
                slot_src[(size_t)wave * MAXE + slot] = b * NN + n + lane;
        }
        base += __popc(m);
        if (base >= MAXE) break;                      /* rest is overflow -> dropped */
    }
    int filled = base < MAXE ? base : MAXE;
    for (int s = filled + lane; s < MAXE; s += 32)
        slot_src[(size_t)wave * MAXE + s] = -1;
}

/* ------------------------------------------------------------------ */
/* Kernel 3: one wave per ROWS=4 consecutive (bin,slot) rows.          */
/* Pipelined CDNA5 async data-mover: issue up to 4 per-lane 16B        */
/* async loads to LDS, drain ASYNCcnt ONCE, then issue the 4 async     */
/* stores (NT hint: output is write-once streaming; keeps the gathered */
/* feature table hot in the 192MB L2). Empty rows write zeroed float4s */
/* directly, so this single pass produces the whole 105 MB output.     */
/* ------------------------------------------------------------------ */
__global__ __launch_bounds__(256) void binner_gather_kernel(
    const float* __restrict__ feat,            /* (B*N, F)      */
    const int* __restrict__ slot_src,          /* (NBINS*MAXE)  */
    float* __restrict__ out_feat,              /* (NBINS*MAXE,F)*/
    float* __restrict__ out_mask,              /* (NBINS*MAXE)  */
    float* __restrict__ out_kb,                /* (NBINS*MAXE)  */
    float* __restrict__ out_kn)                /* (NBINS*MAXE)  */
{
    __shared__ float smem[8 * ROWS * FF];      /* 16 KB: 512 B per row, 4 rows per wave */

    int w    = threadIdx.x >> 5;
    int lane = threadIdx.x & 31;
    long long waveId = (long long)blockIdx.x * 8 + w;
    long long row0   = waveId * ROWS;
    if (row0 >= (long long)NBINS * MAXE) return;

    /* metadata for the wave's 4 contiguous rows: lanes 0..3, one row each */
    if (lane < ROWS) {
        int s = slot_src[row0 + lane];
        out_mask[row0 + lane] = (s >= 0) ? 1.0f : 0.0f;
        out_kb[row0 + lane]   = (s >= 0) ? (float)(s / NN) : -1.0f;
        out_kn[row0 + lane]   = (s >= 0) ? (float)(s % NN) : -1.0f;
    }

    int srcs[ROWS];
#pragma unroll
    for (int r = 0; r < ROWS; ++r)
        srcs[r] = slot_src[row0 + r];          /* wave-uniform -> scalar loads */

    /* stage 1: issue all async gathers into LDS (no intervening waits) */
#pragma unroll
    for (int r = 0; r < ROWS; ++r) {
        if (srcs[r] >= 0) {
            const float* sp = feat + (long long)srcs[r] * FF + lane * 4;
            unsigned lds = (unsigned)(size_t)&smem[(w * ROWS + r) * FF + lane * 4];
            asm volatile("global_load_async_to_lds_b128 %0, %1, off"
                         :: "v"(lds), "v"(sp) : "memory");
        }
    }

    /* single drain: all loads complete (loads retire in order) */
    asm volatile("s_wait_asynccnt 0" ::: "memory");

    /* stage 2: stream rows out (async LDS->global, non-temporal) */
#pragma unroll
    for (int r = 0; r < ROWS; ++r) {
        float* dst = out_feat + (row0 + r) * FF + lane * 4;
        if (srcs[r] >= 0) {
            unsigned lds = (unsigned)(size_t)&smem[(w * ROWS + r) * FF + lane * 4];
            asm volatile("global_store_async_from_lds_b128 %0, %1, off th:TH_STORE_NT"
                         :: "v"(dst), "v"(lds) : "memory");
        } else {
            float4 z = make_float4(0.f, 0.f, 0.f, 0.f);
            *(float4*)dst = z;
        }
    }
    /* s_endpgm performs the implicit final ASYNCcnt drain */
}

/* ------------------------------------------------------------------ */
extern "C" void kernel_launch(void* const* d_in, const int* in_sizes, int n_in,
                              void* d_out, int out_size, void* d_ws, size_t ws_size,
                              hipStream_t stream)
{
    (void)in_sizes; (void)n_in; (void)out_size; (void)ws_size;

    const float*         x_features = (const float*)d_in[0];
    const float*         x_coords   = (const float*)d_in[1];
    const unsigned char* mask       = (const unsigned char*)d_in[2];

    float* out      = (float*)d_out;
    float* out_feat = out;                                         /* 800*256*128 */
    float* out_mask = out + (size_t)NBINS * MAXE * FF;             /* 800*256     */
    float* out_kb   = out_mask + (size_t)NBINS * MAXE;
    float* out_kn   = out_kb   + (size_t)NBINS * MAXE;

    int* gids     = (int*)d_ws;                                    /* B*N ints        */
    int* slot_src = gids + (size_t)BB * NN;                        /* NBINS*MAXE ints */

    binner_gid_kernel<<<(BB * NN + 255) / 256, 256, 0, stream>>>(x_coords, mask, gids);
    binner_scan_kernel<<<NBINS / 8, 256, 0, stream>>>(gids, slot_src);
    binner_gather_kernel<<<(NBINS * MAXE) / (8 * ROWS), 256, 0, stream>>>(
        x_features, slot_src, out_feat, out_mask, out_kb, out_kn);
}